// PointNeXt_56624848831032
// MI455X (gfx1250) — compile-verified
//
#include <hip/hip_runtime.h>
#include <math.h>

// ---------------------------------------------------------------------------
// PointNeXt forward for MI455X (gfx1250, wave32).
//  - all channel-mixing 1x1 convs via V_WMMA_F32_16X16X4_F32 (fp32 matrix pipe)
//    16x64 output strip per wave, branchless inner loop (no exec-mask churn)
//  - GroupNorm as two-pass (atomic f32 stats + apply), gathers fused into the
//    stat/apply passes so the (B,m,S,32) grouped tensor is never materialized
//  - FPS: one workgroup per batch, distance array resident in LDS
// ---------------------------------------------------------------------------

typedef __attribute__((ext_vector_type(2))) float v2f;
typedef __attribute__((ext_vector_type(8))) float v8f;

#define GN_EPS 1e-5f

__device__ __forceinline__ float gelu_f(float x) {
  return 0.5f * x * (1.0f + erff(x * 0.70710678118654752440f));
}

// ---------------------------------------------------------------------------
// split input x (B,N,6) -> pos (B,3,N), feat (B,3,N)
// ---------------------------------------------------------------------------
__global__ void k_split(const float* __restrict__ x, float* __restrict__ pos,
                        float* __restrict__ feat, int B, int N) {
  int i = blockIdx.x * blockDim.x + threadIdx.x;
  if (i >= B * N) return;
  int b = i / N, n = i % N;
  const float* p = x + ((size_t)b * N + n) * 6;
  for (int c = 0; c < 3; ++c) {
    pos [((size_t)b * 3 + c) * N + n] = p[c];
    feat[((size_t)b * 3 + c) * N + n] = p[3 + c];
  }
}

// ---------------------------------------------------------------------------
// farthest point sampling: pos (B,3,N) -> fidx (B,S). One workgroup per batch,
// distance array lives in LDS (N <= 4096 -> 16KB of 320KB WGP LDS).
// ---------------------------------------------------------------------------
__global__ __launch_bounds__(256) void k_fps(const float* __restrict__ pos,
                                             int* __restrict__ fidx,
                                             int B, int N, int S) {
  __shared__ float dist[4096];
  __shared__ float rmax[256];
  __shared__ int   rarg[256];
  __shared__ int   cur;
  int b = blockIdx.x;
  const float* px = pos + (size_t)b * 3 * N;
  const float* py = px + N;
  const float* pz = py + N;
  for (int i = threadIdx.x; i < N; i += blockDim.x) dist[i] = 1e10f;
  if (threadIdx.x == 0) cur = 0;
  __syncthreads();
  for (int it = 0; it < S; ++it) {
    int far = cur;
    float cx = px[far], cy = py[far], cz = pz[far];
    float best = -1.0f; int bestn = 0;
    for (int n = threadIdx.x; n < N; n += blockDim.x) {
      float dx = px[n] - cx, dy = py[n] - cy, dz = pz[n] - cz;
      float d = dx * dx + dy * dy + dz * dz;
      float dd = fminf(dist[n], d);
      dist[n] = dd;
      if (dd > best) { best = dd; bestn = n; }
    }
    rmax[threadIdx.x] = best; rarg[threadIdx.x] = bestn;
    __syncthreads();
    for (int off = 128; off > 0; off >>= 1) {
      if ((int)threadIdx.x < off) {
        float om = rmax[threadIdx.x + off]; int oa = rarg[threadIdx.x + off];
        if (om > rmax[threadIdx.x] ||
            (om == rmax[threadIdx.x] && oa < rarg[threadIdx.x])) {
          rmax[threadIdx.x] = om; rarg[threadIdx.x] = oa;
        }
      }
      __syncthreads();
    }
    if (threadIdx.x == 0) { fidx[b * S + it] = far; cur = rarg[0]; }
    __syncthreads();
  }
}

// gather new_pos (B,3,S) = pos[:, :, fidx]
__global__ void k_gather_pos(const float* __restrict__ pos, const int* __restrict__ fidx,
                             float* __restrict__ newpos, int B, int N, int S) {
  int i = blockIdx.x * blockDim.x + threadIdx.x;
  if (i >= B * S) return;
  int b = i / S, s = i % S;
  int j = fidx[b * S + s];
  for (int c = 0; c < 3; ++c)
    newpos[((size_t)b * 3 + c) * S + s] = pos[((size_t)b * 3 + c) * N + j];
}

// ---------------------------------------------------------------------------
// ball query: first K indices (ascending) with d^2 <= r^2; pad with first hit.
// ---------------------------------------------------------------------------
__global__ void k_ball_query(const float* __restrict__ xyz,      // (B,3,N)
                             const float* __restrict__ new_xyz,  // (B,3,S)
                             int* __restrict__ idx,               // (B,S,K)
                             int B, int N, int S, int K, float r2) {
  int i = blockIdx.x * blockDim.x + threadIdx.x;
  if (i >= B * S) return;
  int b = i / S, s = i % S;
  const float* px = xyz + (size_t)b * 3 * N;
  const float* py = px + N;
  const float* pz = py + N;
  float cx = new_xyz[((size_t)b * 3 + 0) * S + s];
  float cy = new_xyz[((size_t)b * 3 + 1) * S + s];
  float cz = new_xyz[((size_t)b * 3 + 2) * S + s];
  int* out = idx + ((size_t)b * S + s) * K;
  int cnt = 0, first = -1;
  for (int n = 0; n < N && cnt < K; ++n) {
    float dx = px[n] - cx, dy = py[n] - cy, dz = pz[n] - cz;
    if (dx * dx + dy * dy + dz * dz <= r2) {
      if (first < 0) first = n;
      out[cnt++] = n;
    }
  }
  if (first < 0) first = 0;
  for (int k = cnt; k < K; ++k) out[k] = first;
}

// ---------------------------------------------------------------------------
// downsample grouping: feat (B, Cf+3, S*K); channels [0,Cf)=gathered features,
// [Cf,Cf+3)=grouped_pos - new_pos
// ---------------------------------------------------------------------------
__global__ void k_group_ds(const float* __restrict__ xyz,     // (B,3,N)
                           const float* __restrict__ new_xyz, // (B,3,S)
                           const float* __restrict__ xfeat,   // (B,Cf,N)
                           const int* __restrict__ idx,       // (B,S,K)
                           float* __restrict__ feat,          // (B,Cf+3,S*K)
                           int B, int N, int S, int K, int Cf) {
  int i = blockIdx.x * blockDim.x + threadIdx.x;
  if (i >= B * S * K) return;
  int k = i % K; int t = i / K; int s = t % S; int b = t / S;
  int j = idx[((size_t)b * S + s) * K + k];
  int C = Cf + 3;
  size_t SK = (size_t)S * K;
  size_t col = (size_t)s * K + k;
  for (int c = 0; c < Cf; ++c)
    feat[((size_t)b * C + c) * SK + col] = xfeat[((size_t)b * Cf + c) * N + j];
  for (int d = 0; d < 3; ++d)
    feat[((size_t)b * C + Cf + d) * SK + col] =
        xyz[((size_t)b * 3 + d) * N + j] - new_xyz[((size_t)b * 3 + d) * S + s];
}

// ---------------------------------------------------------------------------
// fp32 WMMA GEMM: Y[b,co,n] = sum_ci W[co,ci]*X[b,ci,n] + bias[co]
// Each wave32 computes a 16(M) x 64(N) strip: 4 accumulators share one A
// fetch per K-quad (4 v_wmma per 10 loads). Main loop is fully branchless:
// OOB rows/cols read from clamped (valid) addresses; the garbage lands only
// in D rows/cols that the guarded epilogue never stores. The K tail zeroes
// the A operand with register selects (v_cndmask), never conditional loads.
//
// A 16x4 layout: lanes 0-15 row M=l, {K0,K1}; lanes 16-31 row M=l, {K2,K3}.
// B 4x16 layout (mirrors C half-split): VGPR v, lanes 0-15 -> K=v, lanes
// 16-31 -> K=v+2, N=lane%16.  C/D: VGPR r, M = r + 8*(lane/16), N = lane%16.
// ---------------------------------------------------------------------------
__global__ __launch_bounds__(128) void k_gemm_bias(
    const float* __restrict__ W, const float* __restrict__ X,
    const float* __restrict__ bias, float* __restrict__ Y,
    int Co, int Ci, int N) {
  const int wave = threadIdx.x >> 5;
  const int lane = threadIdx.x & 31;
  const int half = lane >> 4;
  const int l16  = lane & 15;
  const int nbase = (blockIdx.x * 4 + wave) * 64;
  const int co0 = blockIdx.y * 16;
  const int b = blockIdx.z;
  if (nbase >= N) return;  // wave-uniform exit: EXEC stays all-ones for WMMA
  const float* Xb = X + (size_t)b * Ci * N;
  float* Yb = Y + (size_t)b * Co * N;

  // clamped A row (OOB rows never stored)
  const int arow = co0 + l16;
  const float* Wr = W + (size_t)(arow < Co ? arow : Co - 1) * Ci;

  // 4 tile columns; wave-uniform clamps (SALU selects)
  const int n0 = nbase + l16;
  const int n1 = (nbase + 16 < N) ? n0 + 16 : n0;
  const int n2 = (nbase + 32 < N) ? n0 + 32 : n0;
  const int n3 = (nbase + 48 < N) ? n0 + 48 : n0;

  v8f acc0 = {}, acc1 = {}, acc2 = {}, acc3 = {};
  const int Ci4 = Ci & ~3;
  for (int k0 = 0; k0 < Ci4; k0 += 4) {
    const int ak = k0 + half * 2;
    v2f a;
    a.x = Wr[ak];
    a.y = Wr[ak + 1];
    const float* x0 = Xb + (size_t)ak * N;
    const float* x1 = x0 + N;
    v2f b0, b1, b2, b3;
    b0.x = x0[n0]; b0.y = x1[n0];
    b1.x = x0[n1]; b1.y = x1[n1];
    b2.x = x0[n2]; b2.y = x1[n2];
    b3.x = x0[n3]; b3.y = x1[n3];
    if (k0 + 4 < Ci4)  // pull next K-slab toward L0/L2
      __builtin_prefetch(x0 + (size_t)4 * N + n0, 0, 1);
    acc0 = __builtin_amdgcn_wmma_f32_16x16x4_f32(false, a, false, b0, (short)0, acc0, false, false);
    acc1 = __builtin_amdgcn_wmma_f32_16x16x4_f32(false, a, false, b1, (short)0, acc1, false, false);
    acc2 = __builtin_amdgcn_wmma_f32_16x16x4_f32(false, a, false, b2, (short)0, acc2, false, false);
    acc3 = __builtin_amdgcn_wmma_f32_16x16x4_f32(false, a, false, b3, (short)0, acc3, false, false);
  }
  if (Ci4 < Ci) {  // K tail: load from clamped k, zero A via register selects
    const int ak = Ci4 + half * 2;
    const int ak0 = (ak     < Ci) ? ak     : 0;
    const int ak1 = (ak + 1 < Ci) ? ak + 1 : 0;
    float w0 = Wr[ak0], w1 = Wr[ak1];
    v2f a;
    a.x = (ak     < Ci) ? w0 : 0.0f;
    a.y = (ak + 1 < Ci) ? w1 : 0.0f;
    const float* x0 = Xb + (size_t)ak0 * N;
    const float* x1 = Xb + (size_t)ak1 * N;
    v2f b0, b1, b2, b3;
    b0.x = x0[n0]; b0.y = x1[n0];
    b1.x = x0[n1]; b1.y = x1[n1];
    b2.x = x0[n2]; b2.y = x1[n2];
    b3.x = x0[n3]; b3.y = x1[n3];
    acc0 = __builtin_amdgcn_wmma_f32_16x16x4_f32(false, a, false, b0, (short)0, acc0, false, false);
    acc1 = __builtin_amdgcn_wmma_f32_16x16x4_f32(false, a, false, b1, (short)0, acc1, false, false);
    acc2 = __builtin_amdgcn_wmma_f32_16x16x4_f32(false, a, false, b2, (short)0, acc2, false, false);
    acc3 = __builtin_amdgcn_wmma_f32_16x16x4_f32(false, a, false, b3, (short)0, acc3, false, false);
  }

  for (int r = 0; r < 8; ++r) {
    const int co = co0 + r + half * 8;
    if (co >= Co) continue;
    const float bv = bias ? bias[co] : 0.0f;
    float* Yr = Yb + (size_t)co * N;
    /* tile 0 always valid */            Yr[n0]      = acc0[r] + bv;
    if (nbase + 16 < N)                  Yr[nbase + 16 + l16] = acc1[r] + bv;
    if (nbase + 32 < N)                  Yr[nbase + 32 + l16] = acc2[r] + bv;
    if (nbase + 48 < N)                  Yr[nbase + 48 + l16] = acc3[r] + bv;
  }
}

// ---------------------------------------------------------------------------
// GroupNorm two-pass helpers
// ---------------------------------------------------------------------------
__global__ void k_zero(float* __restrict__ p, int n) {
  int i = blockIdx.x * blockDim.x + threadIdx.x;
  if (i < n) p[i] = 0.0f;
}

__device__ __forceinline__ void reduce_pair_atomic(float s, float ss, float* stats, int bg) {
  for (int off = 16; off > 0; off >>= 1) {
    s  += __shfl_down(s, off, 32);
    ss += __shfl_down(ss, off, 32);
  }
  __shared__ float ls[2][8];
  int wave = threadIdx.x >> 5, lane = threadIdx.x & 31;
  if (lane == 0) { ls[0][wave] = s; ls[1][wave] = ss; }
  __syncthreads();
  if (threadIdx.x == 0) {
    float S = 0.0f, SS = 0.0f;
    int nw = blockDim.x >> 5;
    for (int w = 0; w < nw; ++w) { S += ls[0][w]; SS += ls[1][w]; }
    atomicAdd(&stats[bg * 2 + 0], S);
    atomicAdd(&stats[bg * 2 + 1], SS);
  }
}

// stats over X (B,C,N) per (b, group): stats[(b*G+g)*2] = sum, +1 = sumsq
__global__ __launch_bounds__(256) void k_gn_stats(const float* __restrict__ X,
                                                  float* __restrict__ stats,
                                                  int B, int C, int N, int G) {
  int bg = blockIdx.y;
  int b = bg / G, g = bg % G;
  int CperG = C / G;
  size_t base = (size_t)b * C * N + (size_t)g * CperG * N;
  size_t total = (size_t)CperG * N;
  float s = 0.0f, ss = 0.0f;
  for (size_t i = (size_t)blockIdx.x * blockDim.x + threadIdx.x; i < total;
       i += (size_t)gridDim.x * blockDim.x) {
    float v = X[base + i];
    s += v; ss += v * v;
  }
  reduce_pair_atomic(s, ss, stats, bg);
}

// y = [residual +] [gelu] ( gamma*(x-mu)*rsqrt(var+eps) + beta )
__global__ void k_gn_apply(const float* __restrict__ X, float* __restrict__ Y,
                           const float* __restrict__ stats,
                           const float* __restrict__ gamma, const float* __restrict__ beta,
                           const float* __restrict__ residual,
                           int B, int C, int N, int G, int do_gelu) {
  size_t i = (size_t)blockIdx.x * blockDim.x + threadIdx.x;
  size_t tot = (size_t)B * C * N;
  if (i >= tot) return;
  size_t t = i / N;
  int c = (int)(t % C), b = (int)(t / C);
  int g = c / (C / G);
  float cnt = (float)(C / G) * (float)N;
  float mu  = stats[(b * G + g) * 2 + 0] / cnt;
  float var = stats[(b * G + g) * 2 + 1] / cnt - mu * mu;
  float y = (X[i] - mu) * rsqrtf(var + GN_EPS) * gamma[c] + beta[c];
  if (do_gelu) y = gelu_f(y);
  if (residual) y += residual[i];
  Y[i] = y;
}

// max over K: X (B,C,S*K) -> Y (B,C,S)
__global__ void k_max_k(const float* __restrict__ X, float* __restrict__ Y,
                        int B, int C, int S, int K) {
  size_t i = (size_t)blockIdx.x * blockDim.x + threadIdx.x;
  size_t tot = (size_t)B * C * S;
  if (i >= tot) return;
  int s = (int)(i % S); size_t t = i / S;
  int c = (int)(t % C), b = (int)(t / C);
  const float* p = X + ((size_t)(b * C + c) * S) * K + (size_t)s * K;
  float m = p[0];
  for (int k = 1; k < K; ++k) m = fmaxf(m, p[k]);
  Y[i] = m;
}

// ---------------------------------------------------------------------------
// inv-res block: GN over gathered h*dw, fused (never materializes (B,m,S,K)).
// ---------------------------------------------------------------------------
__global__ __launch_bounds__(256) void k_gdw_stats(const float* __restrict__ h,  // (B,m,S)
                                                   const int* __restrict__ idx,  // (B,S,K)
                                                   const float* __restrict__ dw, // (m)
                                                   float* __restrict__ stats,
                                                   int B, int m, int S, int K, int G) {
  int bg = blockIdx.y;
  int b = bg / G, g = bg % G;
  int CperG = m / G;
  size_t total = (size_t)CperG * S * K;
  float s = 0.0f, ss = 0.0f;
  for (size_t t = (size_t)blockIdx.x * blockDim.x + threadIdx.x; t < total;
       t += (size_t)gridDim.x * blockDim.x) {
    int k = (int)(t % K); size_t u = t / K;
    int sp = (int)(u % S);
    int c = g * CperG + (int)(u / S);
    int j = idx[((size_t)b * S + sp) * K + k];
    float v = h[((size_t)b * m + c) * S + j] * dw[c];
    s += v; ss += v * v;
  }
  reduce_pair_atomic(s, ss, stats, bg);
}

// apply pass: out(B,m,S) = gelu( max_k gamma*( h[..,idx]*dw - mu )*rs + beta )
__global__ void k_gdw_apply(const float* __restrict__ h, const int* __restrict__ idx,
                            const float* __restrict__ dw, const float* __restrict__ stats,
                            const float* __restrict__ gamma, const float* __restrict__ beta,
                            float* __restrict__ out, int B, int m, int S, int K, int G) {
  size_t i = (size_t)blockIdx.x * blockDim.x + threadIdx.x;
  size_t tot = (size_t)B * m * S;
  if (i >= tot) return;
  int s = (int)(i % S); size_t t = i / S;
  int c = (int)(t % m), b = (int)(t / m);
  int g = c / (m / G);
  float cnt = (float)(m / G) * (float)S * (float)K;
  float mu  = stats[(b * G + g) * 2 + 0] / cnt;
  float var = stats[(b * G + g) * 2 + 1] / cnt - mu * mu;
  float rs = rsqrtf(var + GN_EPS);
  const int* ip = idx + ((size_t)b * S + s) * K;
  const float* hp = h + ((size_t)b * m + c) * S;
  float dwc = dw[c], ga = gamma[c], be = beta[c];
  float best = -3.402823466e38f;
  for (int k = 0; k < K; ++k) {
    float v = hp[ip[k]] * dwc;
    best = fmaxf(best, (v - mu) * rs * ga + be);
  }
  out[i] = gelu_f(best);
}

// ---------------------------------------------------------------------------
// 3-NN for upsampling + inverse-distance weights
// ---------------------------------------------------------------------------
__global__ void k_knn3(const float* __restrict__ ps,  // (B,3,Ns) skip positions
                       const float* __restrict__ pu,  // (B,3,Nu) coarse positions
                       int* __restrict__ idx3, float* __restrict__ w3,
                       int B, int Ns, int Nu) {
  int i = blockIdx.x * blockDim.x + threadIdx.x;
  if (i >= B * Ns) return;
  int b = i / Ns, s = i % Ns;
  const float* ux = pu + (size_t)b * 3 * Nu;
  const float* uy = ux + Nu;
  const float* uz = uy + Nu;
  float cx = ps[((size_t)b * 3 + 0) * Ns + s];
  float cy = ps[((size_t)b * 3 + 1) * Ns + s];
  float cz = ps[((size_t)b * 3 + 2) * Ns + s];
  float d0 = 1e30f, d1 = 1e30f, d2 = 1e30f;
  int i0 = 0, i1 = 0, i2 = 0;
  for (int n = 0; n < Nu; ++n) {
    float dx = ux[n] - cx, dy = uy[n] - cy, dz = uz[n] - cz;
    float d = dx * dx + dy * dy + dz * dz;
    if (d < d0)      { d2 = d1; i2 = i1; d1 = d0; i1 = i0; d0 = d; i0 = n; }
    else if (d < d1) { d2 = d1; i2 = i1; d1 = d;  i1 = n; }
    else if (d < d2) { d2 = d;  i2 = n; }
  }
  float r0 = 1.0f / (d0 + 1e-8f), r1 = 1.0f / (d1 + 1e-8f), r2 = 1.0f / (d2 + 1e-8f);
  float inv = 1.0f / (r0 + r1 + r2);
  size_t o = ((size_t)b * Ns + s) * 3;
  idx3[o] = i0; idx3[o + 1] = i1; idx3[o + 2] = i2;
  w3[o] = r0 * inv; w3[o + 1] = r1 * inv; w3[o + 2] = r2 * inv;
}

// fused (B,Cs+Cu,Ns): [0,Cs)=x_skip, [Cs,Cs+Cu)=interp(x_up)
__global__ void k_interp_concat(const float* __restrict__ xup, const float* __restrict__ xskip,
                                const int* __restrict__ idx3, const float* __restrict__ w3,
                                float* __restrict__ fused,
                                int B, int Cu, int Cs, int Ns, int Nu) {
  size_t i = (size_t)blockIdx.x * blockDim.x + threadIdx.x;
  int C = Cs + Cu;
  size_t tot = (size_t)B * C * Ns;
  if (i >= tot) return;
  int n = (int)(i % Ns); size_t t = i / Ns;
  int c = (int)(t % C), b = (int)(t / C);
  float v;
  if (c < Cs) {
    v = xskip[((size_t)b * Cs + c) * Ns + n];
  } else {
    size_t o = ((size_t)b * Ns + n) * 3;
    const float* xr = xup + ((size_t)b * Cu + (c - Cs)) * Nu;
    v = w3[o] * xr[idx3[o]] + w3[o + 1] * xr[idx3[o + 1]] + w3[o + 2] * xr[idx3[o + 2]];
  }
  fused[i] = v;
}

// ===========================================================================
// Host side
// ===========================================================================

// Param leaf indexing: d_in[0] = x; params flattened jax-tree style
// (dict keys sorted alphabetically, list in order).
enum {
  IDX_X = 0,
  CLS_B1 = 1, CLS_B2 = 2, CLS_BE1 = 3, CLS_G1 = 4, CLS_W1 = 5, CLS_W2 = 6,
  FP1 = 7, FP2 = 22, FP3 = 37, FP4 = 52,   // each: block(11) + fb,fbe,fg,fw
  S1 = 67, S2 = 82, S3 = 97, S4 = 134,     // each: blocks(11*d) + ds(b,be,g,w)
};
// block leaf offsets (sorted keys): b1,b2,be1,be2,bedw,dw,g1,g2,gdw,w1,w2
enum { BK_B1 = 0, BK_B2, BK_BE1, BK_BE2, BK_BEDW, BK_DW, BK_G1, BK_G2, BK_GDW, BK_W1, BK_W2 };

struct Bump {
  char* base; size_t off; size_t cap;
  void* get(size_t bytes) {
    size_t o = (off + 255) & ~(size_t)255;
    off = o + bytes;
    return base + o;
  }
  float* f(size_t n) { return (float*)get(n * sizeof(float)); }
  int*   i(size_t n) { return (int*)get(n * sizeof(int)); }
};

static inline const float* P(void* const* din, int i) { return (const float*)din[i]; }

static void gemm(hipStream_t st, const float* W, const float* X, const float* bias,
                 float* Y, int B, int Co, int Ci, int N) {
  dim3 g((N + 255) / 256, (Co + 15) / 16, B);
  k_gemm_bias<<<g, 128, 0, st>>>(W, X, bias, Y, Co, Ci, N);
}

static void group_norm(hipStream_t st, Bump& ws, float* X, int B, int C, int N,
                       const float* g, const float* be, int do_gelu, const float* res) {
  float* stats = ws.f((size_t)B * 4 * 2);
  k_zero<<<1, 64, 0, st>>>(stats, B * 4 * 2);
  dim3 sg(48, B * 4);
  k_gn_stats<<<sg, 256, 0, st>>>(X, stats, B, C, N, 4);
  size_t tot = (size_t)B * C * N;
  k_gn_apply<<<(unsigned)((tot + 255) / 256), 256, 0, st>>>(X, X, stats, g, be, res,
                                                            B, C, N, 4, do_gelu);
}

// inv_res_mlp block: x (B,c,S) in, returns new (B,c,S)
static float* run_block(hipStream_t st, Bump& ws, void* const* din, int base,
                        float* x, const float* pos, int B, int S, int c, float radius) {
  const int K = 32, m = 4 * c;
  float* h1 = ws.f((size_t)B * m * S);
  gemm(st, P(din, base + BK_W1), x, P(din, base + BK_B1), h1, B, m, c, S);
  group_norm(st, ws, h1, B, m, S, P(din, base + BK_G1), P(din, base + BK_BE1), 1, nullptr);

  int* idx = ws.i((size_t)B * S * K);
  k_ball_query<<<(B * S + 255) / 256, 256, 0, st>>>(pos, pos, idx, B, S, S, K,
                                                    radius * radius);
  float* stats = ws.f((size_t)B * 4 * 2);
  k_zero<<<1, 64, 0, st>>>(stats, B * 4 * 2);
  dim3 sg(48, B * 4);
  k_gdw_stats<<<sg, 256, 0, st>>>(h1, idx, P(din, base + BK_DW), stats, B, m, S, K, 4);
  float* h2 = ws.f((size_t)B * m * S);
  size_t tot = (size_t)B * m * S;
  k_gdw_apply<<<(unsigned)((tot + 255) / 256), 256, 0, st>>>(
      h1, idx, P(din, base + BK_DW), stats, P(din, base + BK_GDW), P(din, base + BK_BEDW),
      h2, B, m, S, K, 4);

  float* h3 = ws.f((size_t)B * c * S);
  gemm(st, P(din, base + BK_W2), h2, P(din, base + BK_B2), h3, B, c, m, S);
  group_norm(st, ws, h3, B, c, S, P(din, base + BK_G2), P(din, base + BK_BE2), 0, x);
  return h3;
}

static void run_stage(hipStream_t st, Bump& ws, void* const* din, int base, int depth,
                      const float* feat, int Cf, const float* pos, int Nin,
                      int S, float radius, int Cout, float** xOut, float** posOut, int B) {
  const int K = 32;
  int dsb = base + 11 * depth;  // ds leaves: b, be, g, w
  int* fidx = ws.i((size_t)B * S);
  k_fps<<<B, 256, 0, st>>>(pos, fidx, B, Nin, S);
  float* newpos = ws.f((size_t)B * 3 * S);
  k_gather_pos<<<(B * S + 255) / 256, 256, 0, st>>>(pos, fidx, newpos, B, Nin, S);
  int* idx = ws.i((size_t)B * S * K);
  k_ball_query<<<(B * S + 255) / 256, 256, 0, st>>>(pos, newpos, idx, B, Nin, S, K,
                                                    radius * radius);
  int C = Cf + 3;
  size_t SK = (size_t)S * K;
  float* gfeat = ws.f((size_t)B * C * SK);
  k_group_ds<<<(B * S * K + 255) / 256, 256, 0, st>>>(pos, newpos, feat, idx, gfeat,
                                                      B, Nin, S, K, Cf);
  float* conv = ws.f((size_t)B * Cout * SK);
  gemm(st, P(din, dsb + 3), gfeat, P(din, dsb + 0), conv, B, Cout, C, (int)SK);
  group_norm(st, ws, conv, B, Cout, (int)SK, P(din, dsb + 2), P(din, dsb + 1), 1, nullptr);
  float* xm = ws.f((size_t)B * Cout * S);
  size_t tot = (size_t)B * Cout * S;
  k_max_k<<<(unsigned)((tot + 255) / 256), 256, 0, st>>>(conv, xm, B, Cout, S, K);
  float* xcur = xm;
  for (int d = 0; d < depth; ++d)
    xcur = run_block(st, ws, din, base + 11 * d, xcur, newpos, B, S, Cout, radius);
  *xOut = xcur;
  *posOut = newpos;
}

static float* run_upsample(hipStream_t st, Bump& ws, void* const* din, int base,
                           const float* xup, int Cu, const float* posup, int Nu,
                           const float* xskip, int Cs, const float* posskip, int Ns,
                           int Cout, float radius, int B) {
  int* idx3 = ws.i((size_t)B * Ns * 3);
  float* w3 = ws.f((size_t)B * Ns * 3);
  k_knn3<<<(B * Ns + 255) / 256, 256, 0, st>>>(posskip, posup, idx3, w3, B, Ns, Nu);
  int Cin = Cs + Cu;
  float* fused = ws.f((size_t)B * Cin * Ns);
  size_t tot = (size_t)B * Cin * Ns;
  k_interp_concat<<<(unsigned)((tot + 255) / 256), 256, 0, st>>>(xup, xskip, idx3, w3,
                                                                 fused, B, Cu, Cs, Ns, Nu);
  float* h = ws.f((size_t)B * Cout * Ns);
  gemm(st, P(din, base + 14), fused, P(din, base + 11), h, B, Cout, Cin, Ns);
  group_norm(st, ws, h, B, Cout, Ns, P(din, base + 13), P(din, base + 12), 1, nullptr);
  return run_block(st, ws, din, base + 0, h, posskip, B, Ns, Cout, radius);
}

extern "C" void kernel_launch(void* const* d_in, const int* in_sizes, int n_in,
                              void* d_out, int out_size, void* d_ws, size_t ws_size,
                              hipStream_t stream) {
  (void)in_sizes; (void)n_in; (void)out_size;
  const int B = 2, N0 = 4096;
  Bump ws{(char*)d_ws, 0, ws_size};

  const float* x = (const float*)d_in[IDX_X];
  float* pos0  = ws.f((size_t)B * 3 * N0);
  float* feat0 = ws.f((size_t)B * 3 * N0);
  k_split<<<(B * N0 + 255) / 256, 256, 0, stream>>>(x, pos0, feat0, B, N0);

  float *l1x, *l1p, *l2x, *l2p, *l3x, *l3p, *l4x, *l4p;
  run_stage(stream, ws, d_in, S1, 1, feat0, 3,   pos0, N0,   1024, 0.10f, 32,  &l1x, &l1p, B);
  run_stage(stream, ws, d_in, S2, 1, l1x,  32,  l1p, 1024,  512, 0.15f, 64,  &l2x, &l2p, B);
  run_stage(stream, ws, d_in, S3, 3, l2x,  64,  l2p,  512,  256, 0.20f, 128, &l3x, &l3p, B);
  run_stage(stream, ws, d_in, S4, 1, l3x, 128,  l3p,  256,  128, 0.30f, 256, &l4x, &l4p, B);

  float* u3 = run_upsample(stream, ws, d_in, FP4, l4x, 256, l4p, 128,
                           l3x, 128, l3p, 256, 128, 0.30f, B);
  float* u2 = run_upsample(stream, ws, d_in, FP3, u3, 128, l3p, 256,
                           l2x, 64, l2p, 512, 64, 0.20f, B);
  float* u1 = run_upsample(stream, ws, d_in, FP2, u2, 64, l2p, 512,
                           l1x, 32, l1p, 1024, 32, 0.15f, B);
  float* u0 = run_upsample(stream, ws, d_in, FP1, u1, 32, l1p, 1024,
                           feat0, 3, pos0, N0, 32, 0.10f, B);

  // classification head: conv(32,32) + GN + GELU + conv(2,32) -> d_out (B,2,N0)
  float* h = ws.f((size_t)B * 32 * N0);
  gemm(stream, P(d_in, CLS_W1), u0, P(d_in, CLS_B1), h, B, 32, 32, N0);
  group_norm(stream, ws, h, B, 32, N0, P(d_in, CLS_G1), P(d_in, CLS_BE1), 1, nullptr);
  gemm(stream, P(d_in, CLS_W2), h, P(d_in, CLS_B2), (float*)d_out, B, 2, 32, N0);
}